// AEVComputer_4922032522007
// MI455X (gfx1250) — compile-verified
//
#include <hip/hip_runtime.h>
#include <hip/hip_bf16.h>
#include <math.h>

typedef __attribute__((ext_vector_type(2))) float v2f;
typedef __attribute__((ext_vector_type(8))) float v8f;

#define NA    48  // atoms per molecule
#define NS    4   // species
#define NSHR  16  // radial shells
#define NSHA  4   // angular radial shells
#define NSHZ  8   // angular sections
#define NPAIR 10  // S*(S+1)/2
#define RAD_LEN (NS*NSHR)          // 64
#define ANG_LEN (NPAIR*NSHA*NSHZ)  // 320
#define AEV_LEN (RAD_LEN+ANG_LEN)  // 384
#define NWAVE 4
#define BLOCK (NWAVE*32)

__global__ __launch_bounds__(BLOCK) void aev_kernel(
    const int*   __restrict__ species,   // [B,A]
    const float* __restrict__ coords,    // [B,A,3]
    const float* __restrict__ etaR_p,    // [1]
    const float* __restrict__ shfR,      // [16]
    const float* __restrict__ etaA_p,    // [1]
    const float* __restrict__ zeta_p,    // [1]
    const float* __restrict__ shfA,      // [4]
    const float* __restrict__ shfZ,      // [8]
    const int*   __restrict__ triu,      // [S,S]
    const float* __restrict__ rcr_p,     // [1]
    const float* __restrict__ rca_p,     // [1]
    float*       __restrict__ out)       // [B,A,384]
{
    const int bi   = blockIdx.x;         // b*NA + i
    const int b    = bi / NA;
    const int i    = bi % NA;
    const int tid  = threadIdx.x;        // 0..127
    const int lane = tid & 31;
    const int wv   = tid >> 5;           // wave id 0..3

    __shared__ float s_px[NA], s_py[NA], s_pz[NA];
    __shared__ float s_vx[NA], s_vy[NA], s_vz[NA];
    __shared__ float s_dist[NA], s_fca[NA], s_fcr[NA];
    __shared__ int   s_spec[NA];
    __shared__ float s_shfr[NSHR];
    __shared__ int   s_triu[NS*NS];
    __shared__ float s_rt[NA*NSHR];           // rt[j][r], WMMA B-operand staging
    __shared__ float s_ang[NWAVE*ANG_LEN];    // per-wave angular accumulators

    const float etaR = etaR_p[0];
    const float etaA = etaA_p[0];
    const float zeta = zeta_p[0];
    const float Rcr  = rcr_p[0];
    const float Rca  = rca_p[0];
    const float PI_F = 3.14159265358979323846f;

    // ---- stage batch data into LDS ----
    if (tid < NA) {
        const int gi = b * NA + tid;
        s_px[tid] = coords[gi * 3 + 0];
        s_py[tid] = coords[gi * 3 + 1];
        s_pz[tid] = coords[gi * 3 + 2];
        s_spec[tid] = species[gi];
    }
    if (tid >= 64 && tid < 64 + NSHR)        s_shfr[tid - 64] = shfR[tid - 64];
    if (tid >= 96 && tid < 96 + NS*NS)       s_triu[tid - 96] = triu[tid - 96];
    for (int t = tid; t < NWAVE*ANG_LEN; t += BLOCK) s_ang[t] = 0.0f;
    __syncthreads();

    const bool  validI = (s_spec[i] >= 0);
    const float pix = s_px[i], piy = s_py[i], piz = s_pz[i];

    // ---- per-neighbor geometry: vec, dist, cutoffs ----
    if (tid < NA) {
        const int j = tid;
        const float vx = s_px[j] - pix;
        const float vy = s_py[j] - piy;
        const float vz = s_pz[j] - piz;
        const float d  = sqrtf(vx * vx + vy * vy + vz * vz);
        const bool pv  = validI && (s_spec[j] >= 0) && (j != i);
        float fcr = 0.0f, fca = 0.0f;
        if (pv && d <= Rcr) fcr = 0.5f * __cosf(PI_F * d / Rcr) + 0.5f;
        if (pv && d <= Rca) fca = 0.5f * __cosf(PI_F * d / Rca) + 0.5f;
        s_vx[j] = vx; s_vy[j] = vy; s_vz[j] = vz;
        s_dist[j] = d; s_fcr[j] = fcr; s_fca[j] = fca;
    }
    __syncthreads();

    // ---- radial terms rt[j][r] = 0.25*exp(-etaR*(d-ShfR)^2)*fc ----
    for (int t = tid; t < NA * NSHR; t += BLOCK) {
        const int j = t >> 4;
        const int r = t & 15;
        const float diff = s_dist[j] - s_shfr[r];
        s_rt[t] = 0.25f * __expf(-etaR * diff * diff) * s_fcr[j];
    }
    __syncthreads();

    // ---- radial species contraction (wave 0): chained V_WMMA_F32_16X16X4_F32 ----
    // D[s][r] += A[s][k]*B[k][r]; A = one-hot(species) 16x4, B = rt chunk 4x16, K=48.
    // A layout: lanes 0-15 -> {K=0,K=1}, lanes 16-31 -> {K=2,K=3}; M = lane&15.
    if (wv == 0) {
        const int m  = lane & 15;          // output row (species)
        const int k0 = (lane >> 4) << 1;   // K offset for this half-wave
        v8f acc = {};
        #pragma unroll
        for (int c = 0; c < NA / 4; ++c) {
            const int jb = c * 4;
            v2f a, bb;
            a.x  = (s_spec[jb + k0 + 0] == m) ? 1.0f : 0.0f;
            a.y  = (s_spec[jb + k0 + 1] == m) ? 1.0f : 0.0f;
            bb.x = s_rt[(jb + k0 + 0) * NSHR + m];
            bb.y = s_rt[(jb + k0 + 1) * NSHR + m];
            acc = __builtin_amdgcn_wmma_f32_16x16x4_f32(
                false, a, false, bb, (short)0, acc, false, false);
        }
        // lanes 0-15 hold rows M=0..7 in acc[0..7]; species rows are M=0..3.
        const long obase = (long)bi * AEV_LEN;
        if (lane < 16) {
            #pragma unroll
            for (int s = 0; s < NS; ++s)
                out[obase + s * NSHR + lane] = acc[s];
        }
    }

    // ---- angular part: lanes = 32 features (a = shfA*8 + shfZ) ----
    // wave w owns outer rows j ≡ w (mod 4): fixed partition -> deterministic order.
    const float myShfA = shfA[lane >> 3];
    const float zval   = shfZ[lane & 7];
    const float cz = __cosf(zval);
    const float sz = __sinf(zval);
    float* myAng = &s_ang[wv * ANG_LEN];

    for (int j = wv; j < NA - 1; j += NWAVE) {
        const float fj = s_fca[j];
        if (fj == 0.0f) continue;                 // wave-uniform skip
        const float dj = s_dist[j];
        const float xj = s_vx[j], yj = s_vy[j], zj = s_vz[j];
        const int   sj = (s_spec[j] < 0) ? 0 : s_spec[j];
        for (int k = j + 1; k < NA; ++k) {
            const float fprod = fj * s_fca[k];
            if (fprod == 0.0f) continue;          // wave-uniform skip
            const float dk  = s_dist[k];
            const float dot = xj * s_vx[k] + yj * s_vy[k] + zj * s_vz[k];
            const float dd  = fmaxf(dj * dk, 1e-8f);
            const float cosa = 0.95f * dot / dd;
            const float sina = sqrtf(fmaxf(1.0f - cosa * cosa, 0.0f));
            const float dsum = 0.5f * (dj + dk);
            const int   sk = (s_spec[k] < 0) ? 0 : s_spec[k];
            const int   p  = s_triu[sj * NS + sk];
            // cos(theta - ShfZ) = cosa*cz + sina*sz  (no acos/cos in the hot loop)
            const float ct = fmaxf(0.5f * (1.0f + cosa * cz + sina * sz), 0.0f);
            const float f1 = powf(ct, zeta);
            const float e  = dsum - myShfA;
            const float f2 = __expf(-etaA * e * e);
            // lane-unique address within this wave's bank: ds_add_f32, no return
            atomicAdd(&myAng[p * 32 + lane], 2.0f * f1 * f2 * fprod);
        }
    }
    __syncthreads();

    // ---- fixed-order cross-wave reduce + write angular features ----
    {
        const long obase = (long)bi * AEV_LEN + RAD_LEN;
        for (int t = tid; t < ANG_LEN; t += BLOCK) {
            float v = s_ang[t];
            v += s_ang[t + ANG_LEN];
            v += s_ang[t + 2 * ANG_LEN];
            v += s_ang[t + 3 * ANG_LEN];
            out[obase + t] = v;
        }
    }
}

extern "C" void kernel_launch(void* const* d_in, const int* in_sizes, int n_in,
                              void* d_out, int out_size, void* d_ws, size_t ws_size,
                              hipStream_t stream) {
    const int*   species = (const int*)  d_in[0];
    const float* coords  = (const float*)d_in[1];
    const float* etaR    = (const float*)d_in[2];
    const float* shfR    = (const float*)d_in[3];
    const float* etaA    = (const float*)d_in[4];
    const float* zeta    = (const float*)d_in[5];
    const float* shfA    = (const float*)d_in[6];
    const float* shfZ    = (const float*)d_in[7];
    const int*   triu    = (const int*)  d_in[8];
    const float* rcr     = (const float*)d_in[9];
    const float* rca     = (const float*)d_in[10];
    float* out = (float*)d_out;

    const int n_atoms = in_sizes[0];            // B*A = 384
    aev_kernel<<<n_atoms, BLOCK, 0, stream>>>(species, coords, etaR, shfR, etaA,
                                              zeta, shfA, shfZ, triu, rcr, rca, out);
}